// ReprojectMetric_64819646431966
// MI455X (gfx1250) — compile-verified
//
#include <hip/hip_runtime.h>
#include <hip/hip_bf16.h>
#include <math.h>

// ---------------------------------------------------------------------------
// ReprojectMetric for MI455X (gfx1250, wave32).
// Problem: B=1, C=64, H=W=512.  L2-resident gather problem; channel reduction
// mapped onto V_WMMA_F32_16X16X4_F32 (f32 in / f32 accumulate, exact-precision
// matrix path), 16 pixels per wave, diagonal harvest.
// ---------------------------------------------------------------------------

#define HH   512
#define WW   512
#define CC   64
#define HWPX (HH * WW)
#define NWAVES (HWPX / 16)          // 16384 per-wave partials
#define CYCLE_THR 0.0025f
#define EPS_Z 1e-06f

typedef __attribute__((ext_vector_type(2))) float v2f;
typedef __attribute__((ext_vector_type(8))) float v8f;

// ---------------------------- helpers --------------------------------------

__device__ inline void inv3x3(const float* K, float* Ki) {
    float a = K[0], b = K[1], c = K[2];
    float d = K[3], e = K[4], f = K[5];
    float g = K[6], h = K[7], i = K[8];
    float A =  (e * i - f * h);
    float B = -(d * i - f * g);
    float Cc = (d * h - e * g);
    float det = a * A + b * B + c * Cc;
    float id = 1.0f / det;
    Ki[0] = A * id;
    Ki[1] = -(b * i - c * h) * id;
    Ki[2] =  (b * f - c * e) * id;
    Ki[3] = B * id;
    Ki[4] =  (a * i - c * g) * id;
    Ki[5] = -(a * f - c * d) * id;
    Ki[6] = Cc * id;
    Ki[7] = -(a * h - b * g) * id;
    Ki[8] =  (a * e - b * d) * id;
}

// unproject (px,py,d) in 'from' cam, transform to 'to' cam, project, -> [-1,1]^2
__device__ inline float2 reproj_pt(float px, float py, float d,
                                   const float* Ki,          // inv(from_K)
                                   const float* cf,          // from c2w [3x4]
                                   const float* ct,          // to   c2w [3x4]
                                   const float* Kt) {        // to_K
    float cx = (Ki[0] * px + Ki[1] * py + Ki[2]) * d;
    float cy = (Ki[3] * px + Ki[4] * py + Ki[5]) * d;
    float cz = (Ki[6] * px + Ki[7] * py + Ki[8]) * d;
    // world = R_f @ cam + t_f   (c2w row-major [r*4 + c], t = col 3)
    float wx = cf[0] * cx + cf[1] * cy + cf[2]  * cz + cf[3];
    float wy = cf[4] * cx + cf[5] * cy + cf[6]  * cz + cf[7];
    float wz = cf[8] * cx + cf[9] * cy + cf[10] * cz + cf[11];
    // q = world - t_t ; cam2 = R_t^T @ q
    float qx = wx - ct[3], qy = wy - ct[7], qz = wz - ct[11];
    float ax = ct[0] * qx + ct[4] * qy + ct[8]  * qz;
    float ay = ct[1] * qx + ct[5] * qy + ct[9]  * qz;
    float az = ct[2] * qx + ct[6] * qy + ct[10] * qz;
    float prx = Kt[0] * ax + Kt[1] * ay + Kt[2] * az;
    float pry = Kt[3] * ax + Kt[4] * ay + Kt[5] * az;
    float prz = Kt[6] * ax + Kt[7] * ay + Kt[8] * az;
    float zs = (fabsf(prz) > EPS_Z) ? prz : EPS_Z;
    float u = prx / zs, v = pry / zs;
    float2 r;
    r.x = u * (2.0f / WW) - 1.0f;
    r.y = v * (2.0f / HH) - 1.0f;
    return r;
}

struct GS {
    int   off[4];
    float w[4];
};

// grid_sample prep: bilinear, align_corners=False, zero padding.
__device__ inline GS gs_make(float gx, float gy) {
    float x = (gx + 1.0f) * (0.5f * WW) - 0.5f;
    float y = (gy + 1.0f) * (0.5f * HH) - 0.5f;
    float fx = floorf(x), fy = floorf(y);
    float wx = x - fx, wy = y - fy;
    int x0 = (int)fx, y0 = (int)fy;
    GS g;
#pragma unroll
    for (int k = 0; k < 4; ++k) {
        int xi = x0 + (k & 1);
        int yi = y0 + (k >> 1);
        bool valid = (xi >= 0) && (xi < WW) && (yi >= 0) && (yi < HH);
        int xc = min(max(xi, 0), WW - 1);
        int yc = min(max(yi, 0), HH - 1);
        g.off[k] = yc * WW + xc;
        float wk = ((k & 1) ? wx : (1.0f - wx)) * ((k >> 1) ? wy : (1.0f - wy));
        g.w[k] = valid ? wk : 0.0f;   // select, no EXEC divergence
    }
    return g;
}

__device__ inline float gs_sample(const float* __restrict__ plane, const GS& g) {
    return g.w[0] * plane[g.off[0]] + g.w[1] * plane[g.off[1]] +
           g.w[2] * plane[g.off[2]] + g.w[3] * plane[g.off[3]];
}

// -------------------- kernel A: reprojection pass --------------------------

__global__ void reproj_kernel(const float* __restrict__ depth1,
                              const float* __restrict__ depth2,
                              const float* __restrict__ K1,
                              const float* __restrict__ K2,
                              const float* __restrict__ c2w1,
                              const float* __restrict__ c2w2,
                              float2* __restrict__ l12,
                              float2* __restrict__ l21) {
    int p = blockIdx.x * blockDim.x + threadIdx.x;
    if (p >= HWPX) return;
    int i = p >> 9;           // p / 512
    int j = p & (WW - 1);     // p % 512
    float px = (float)j + 0.5f;
    float py = (float)i + 0.5f;

    float K1i[9], K2i[9];
    inv3x3(K1, K1i);
    inv3x3(K2, K2i);

    // l1_2 = reproject(l1, depth1, K1, c2w1 -> K2, c2w2)
    l12[p] = reproj_pt(px, py, depth1[p], K1i, c2w1, c2w2, K2);
    // l2_1 = reproject(l2, depth2, K2, c2w2 -> K1, c2w1)
    l21[p] = reproj_pt(px, py, depth2[p], K2i, c2w2, c2w1, K1);
}

// -------------------- kernel B: fused gather + WMMA reduce -----------------
// One wave = 16 consecutive pixels. lane: m = lane%16 (pixel), h = lane/16
// (channel-half).  Per WMMA step t: lane supplies channels {4t+2h, 4t+2h+1}.
// A = w*diff, B = diff  =>  D[m][m] += w[m] * sum_k diff[m][k]^2.

__global__ void __launch_bounds__(256)
loss_kernel(const float* __restrict__ features1,
            const float* __restrict__ features2,
            const float* __restrict__ mask1,
            const float2* __restrict__ l12,
            const float2* __restrict__ l21,
            float* __restrict__ partials) {
    const int lane = threadIdx.x & 31;
    const int wave = threadIdx.x >> 5;
    const int gw   = blockIdx.x * (blockDim.x >> 5) + wave;   // global wave id
    const int m    = lane & 15;
    const int h    = lane >> 4;
    const int p    = gw * 16 + m;                              // this lane's pixel

    const int i = p >> 9;
    const int j = p & (WW - 1);
    // identity grid l1 (exact in f32)
    const float gx = ((float)j + 0.5f) * (2.0f / WW) - 1.0f;
    const float gy = ((float)i + 0.5f) * (2.0f / HH) - 1.0f;

    const float2 c12 = l12[p];
    const float2 c21 = l21[p];

    // l1_2_1 = bilinear sample of the l2_1 field at l1_2
    GS gcyc = gs_make(c12.x, c12.y);
    float sx = 0.0f, sy = 0.0f;
#pragma unroll
    for (int k = 0; k < 4; ++k) {
        float2 v = l21[gcyc.off[k]];
        sx += gcyc.w[k] * v.x;
        sy += gcyc.w[k] * v.y;
    }
    float dx = gx - sx, dy = gy - sy;
    float dist = sqrtf(dx * dx + dy * dy);
    float w = (dist < CYCLE_THR) ? mask1[p] : 0.0f;

    // sampling footprints for the two feature gathers
    GS g1 = gs_make(c12.x, c12.y);   // features2 sampled at l1_2
    GS g2 = gs_make(c21.x, c21.y);   // features1 sampled at l2_1

    v8f acc = {};
#pragma unroll
    for (int t = 0; t < 16; ++t) {
        const int c0 = 4 * t + 2 * h;
        const int c1 = c0 + 1;
        const float* f1p0 = features1 + c0 * HWPX;
        const float* f1p1 = features1 + c1 * HWPX;
        const float* f2p0 = features2 + c0 * HWPX;
        const float* f2p1 = features2 + c1 * HWPX;

        // term 1: f1 - grid_sample(features2, l1_2)
        float d0 = f1p0[p] - gs_sample(f2p0, g1);
        float d1 = f1p1[p] - gs_sample(f2p1, g1);
        {
            v2f a = { w * d0, w * d1 };
            v2f b = { d0, d1 };
            acc = __builtin_amdgcn_wmma_f32_16x16x4_f32(
                false, a, false, b, (short)0, acc, false, false);
        }
        // term 2: f2 - grid_sample(features1, l2_1)
        float e0 = f2p0[p] - gs_sample(f1p0, g2);
        float e1 = f2p1[p] - gs_sample(f1p1, g2);
        {
            v2f a = { w * e0, w * e1 };
            v2f b = { e0, e1 };
            acc = __builtin_amdgcn_wmma_f32_16x16x4_f32(
                false, a, false, b, (short)0, acc, false, false);
        }
    }

    // Harvest diagonal D[m][m]:
    //   m in 0..7  -> VGPR m,   lane m         (lanes 0..7)
    //   m in 8..15 -> VGPR m-8, lane m+16      (lanes 24..31)
    int sel = (lane < 8) ? lane : ((lane >= 24) ? (lane - 24) : -1);
    float tv = 0.0f;
#pragma unroll
    for (int r = 0; r < 8; ++r) tv = (sel == r) ? acc[r] : tv;

    // wave32 reduction (deterministic butterfly)
#pragma unroll
    for (int off = 16; off > 0; off >>= 1)
        tv += __shfl_xor(tv, off, 32);

    if (lane == 0) partials[gw] = tv;
}

// -------------------- kernel C: deterministic final reduce -----------------

__global__ void finalize_kernel(const float* __restrict__ partials,
                                float* __restrict__ out) {
    __shared__ float sm[256];
    float s = 0.0f;
    for (int idx = threadIdx.x; idx < NWAVES; idx += 256)
        s += partials[idx];
    sm[threadIdx.x] = s;
    __syncthreads();
#pragma unroll
    for (int stride = 128; stride > 0; stride >>= 1) {
        if (threadIdx.x < stride) sm[threadIdx.x] += sm[threadIdx.x + stride];
        __syncthreads();
    }
    if (threadIdx.x == 0)
        out[0] = sm[0] * (1.0f / (float)HWPX);   // LOSSFEATMULT == 1
}

// ---------------------------------------------------------------------------

extern "C" void kernel_launch(void* const* d_in, const int* in_sizes, int n_in,
                              void* d_out, int out_size, void* d_ws, size_t ws_size,
                              hipStream_t stream) {
    const float* features1 = (const float*)d_in[0];
    const float* features2 = (const float*)d_in[1];
    const float* depth1    = (const float*)d_in[4];
    const float* depth2    = (const float*)d_in[5];
    const float* mask1     = (const float*)d_in[6];
    const float* K1        = (const float*)d_in[8];
    const float* K2        = (const float*)d_in[9];
    const float* c2w1      = (const float*)d_in[10];
    const float* c2w2      = (const float*)d_in[11];

    float2* l12      = (float2*)d_ws;                 // HWPX float2 (2 MB)
    float2* l21      = l12 + HWPX;                    // HWPX float2 (2 MB)
    float*  partials = (float*)(l21 + HWPX);          // NWAVES floats (64 KB)

    reproj_kernel<<<HWPX / 256, 256, 0, stream>>>(
        depth1, depth2, K1, K2, c2w1, c2w2, l12, l21);

    // 2048 blocks * 8 waves * 16 pixels = 262144 pixels
    loss_kernel<<<NWAVES / 8, 256, 0, stream>>>(
        features1, features2, mask1, l12, l21, partials);

    finalize_kernel<<<1, 256, 0, stream>>>(partials, (float*)d_out);
}